// PrototypeEMA_17849884082283
// MI455X (gfx1250) — compile-verified
//
#include <hip/hip_runtime.h>
#include <hip/hip_bf16.h>

// PrototypeEMA for MI455X (gfx1250, wave32).
//
// Pipeline:
//   0) hipMemsetAsync: zero seg_sum[C,D] + cnt[C] workspace (1.03 MB, L2-resident)
//   1) seg_accum_kernel: stream z (256 MB) once; per-WG 128KB LDS accumulator
//      (1000 classes x 32-dim slice) updated with ds_add_f32 atomics; single
//      global_atomic_add_f32 flush per WG. 8 dim-slices x 64 N-partitions.
//   2) finalize_kernel: per 16-class tile, compute ||mean|| and ||0.99*p+0.01*zc||
//      via V_WMMA_F32_16X16X4_F32 Gram diagonals (A == B register-identical for
//      the f32 WMMA layouts since B = A^T), then EMA/select writes.

#define C_CLASSES 1000
#define DIMS      256
#define MOM       0.99f
#define MOM1      0.01f
#define F_EPS     1e-12f

#define SLICES    8          // dim slices of 32 (one 128B cacheline per sample)
#define SLICE_D   32
#define PARTS     64         // N partitions
#define TPB       256

typedef __attribute__((ext_vector_type(2))) float v2f;
typedef __attribute__((ext_vector_type(8))) float v8f;

__device__ __forceinline__ void lds_fadd(float* p, float v) {
  (void)__hip_atomic_fetch_add(p, v, __ATOMIC_RELAXED, __HIP_MEMORY_SCOPE_WORKGROUP);
}

// ---------------------------------------------------------------------------
// Kernel 1: segment scatter-accumulate into workspace via LDS privatization.
// grid = SLICES * PARTS blocks, TPB threads. 4 lanes per sample row-slice.
// ---------------------------------------------------------------------------
__global__ __launch_bounds__(TPB) void seg_accum_kernel(
    const float* __restrict__ z, const int* __restrict__ y,
    float* __restrict__ seg, float* __restrict__ cnt, int n)
{
  // 1000*32 slice accumulator + 1000 count accumulator = 132 KB (CDNA5 LDS)
  __shared__ float tile[C_CLASSES * SLICE_D + C_CLASSES];
  float* ctile = tile + C_CLASSES * SLICE_D;

  const int tid   = threadIdx.x;
  const int slice = blockIdx.x & (SLICES - 1);
  const int part  = blockIdx.x >> 3;

  for (int i = tid; i < C_CLASSES * SLICE_D + C_CLASSES; i += TPB) tile[i] = 0.0f;
  __syncthreads();

  const int  d0      = slice * SLICE_D;
  const int  sub     = (tid & 3) * 8;          // quad covers 32 dims = 128B line
  const int  spp     = (n + PARTS - 1) / PARTS;
  const int  i0      = part * spp;
  const int  i1      = (i0 + spp < n) ? (i0 + spp) : n;
  const bool cntLane = (slice == 0) && ((tid & 3) == 0);

  for (int i = i0 + (tid >> 2); i < i1; i += TPB / 4) {
    // keep the HBM stream ahead (global_prefetch_b8)
    __builtin_prefetch(z + (size_t)(i + 256) * DIMS + d0 + sub, 0, 1);

    int c = y[i];
    c = (c < 0) ? 0 : ((c >= C_CLASSES) ? (C_CLASSES - 1) : c);

    const float4* zp = (const float4*)(z + (size_t)i * DIMS + d0 + sub);
    float4 a = zp[0];
    float4 b = zp[1];

    float* t = tile + c * SLICE_D + sub;
    lds_fadd(t + 0, a.x); lds_fadd(t + 1, a.y);
    lds_fadd(t + 2, a.z); lds_fadd(t + 3, a.w);
    lds_fadd(t + 4, b.x); lds_fadd(t + 5, b.y);
    lds_fadd(t + 6, b.z); lds_fadd(t + 7, b.w);

    if (cntLane) lds_fadd(ctile + c, 1.0f);
  }
  __syncthreads();

  // Flush LDS accumulator to L2-resident global accumulator.
  for (int idx = tid; idx < C_CLASSES * SLICE_D; idx += TPB) {
    float v = tile[idx];
    if (v != 0.0f) {
      int c = idx >> 5;
      int d = idx & 31;
      unsafeAtomicAdd(seg + (size_t)c * DIMS + d0 + d, v);
    }
  }
  if (slice == 0) {
    for (int c = tid; c < C_CLASSES; c += TPB) {
      float v = ctile[c];
      if (v != 0.0f) unsafeAtomicAdd(cnt + c, v);
    }
  }
}

// ---------------------------------------------------------------------------
// Kernel 2: finalize. One wave per 16-class tile; L2 norms via WMMA f32
// Gram diagonals (A and B=A^T are the same registers under the f32 layouts).
// ---------------------------------------------------------------------------
__global__ __launch_bounds__(256) void finalize_kernel(
    const float* __restrict__ seg, const float* __restrict__ cnt,
    const float* __restrict__ protos, const unsigned char* __restrict__ init_mask,
    float* __restrict__ out_protos, float* __restrict__ out_mask)
{
  __shared__ float normbuf[8 * 16];

  const int  lane = threadIdx.x & 31;     // wave32
  const int  w    = threadIdx.x >> 5;
  const int  clsl = lane & 15;
  const int  c    = blockIdx.x * 128 + w * 16 + clsl;
  const int  koff = (lane < 16) ? 0 : 2;  // f32 WMMA A-layout: hi lanes hold K+2,K+3
  const bool valid = (c < C_CLASSES);
  const int  cc   = valid ? c : (C_CLASSES - 1);

  const float n_f     = valid ? cnt[cc] : 0.0f;
  const bool  present = n_f > 0.0f;
  const float invc    = 1.0f / fmaxf(n_f, 1.0f);
  const float* segrow = seg    + (size_t)cc * DIMS;
  const float* prow   = protos + (size_t)cc * DIMS;

  // ---- pass 1: ||mean||^2 for 16 classes at once via Gram diagonal ----
  v8f acc = {};
  for (int k = 0; k < DIMS; k += 4) {
    v2f m;
    m.x = segrow[k + koff + 0] * invc;
    m.y = segrow[k + koff + 1] * invc;
    if (!valid) { m.x = 0.0f; m.y = 0.0f; }
    acc = __builtin_amdgcn_wmma_f32_16x16x4_f32(
        false, m, false, m, (short)0, acc, false, false);
  }
#pragma unroll
  for (int r = 0; r < 8; ++r) {           // diag(c,c): c<8 -> vgpr c lane c; c>=8 -> vgpr c-8 lane c+16
    if (lane == r)      normbuf[w * 16 + r]     = acc[r];
    if (lane == r + 24) normbuf[w * 16 + r + 8] = acc[r];
  }
  __syncthreads();
  const float inv1 = 1.0f / fmaxf(sqrtf(normbuf[w * 16 + clsl]), F_EPS);
  __syncthreads();

  // ---- pass 2: ||0.99*proto + 0.01*zc||^2 via Gram diagonal ----
  v8f acc2 = {};
  for (int k = 0; k < DIMS; k += 4) {
    float zc0 = segrow[k + koff + 0] * invc * inv1;
    float zc1 = segrow[k + koff + 1] * invc * inv1;
    v2f e;
    e.x = MOM * prow[k + koff + 0] + MOM1 * zc0;
    e.y = MOM * prow[k + koff + 1] + MOM1 * zc1;
    if (!valid) { e.x = 0.0f; e.y = 0.0f; }
    acc2 = __builtin_amdgcn_wmma_f32_16x16x4_f32(
        false, e, false, e, (short)0, acc2, false, false);
  }
#pragma unroll
  for (int r = 0; r < 8; ++r) {
    if (lane == r)      normbuf[w * 16 + r]     = acc2[r];
    if (lane == r + 24) normbuf[w * 16 + r + 8] = acc2[r];
  }
  __syncthreads();
  const float inv2 = 1.0f / fmaxf(sqrtf(normbuf[w * 16 + clsl]), F_EPS);

  // ---- pass 3: EMA / init / present selects, write outputs ----
  const bool init = valid ? (init_mask[cc] != 0) : false;
  for (int k = 0; k < DIMS; k += 4) {
    float p0  = prow[k + koff + 0], p1 = prow[k + koff + 1];
    float zc0 = segrow[k + koff + 0] * invc * inv1;
    float zc1 = segrow[k + koff + 1] * invc * inv1;
    float e0  = (MOM * p0 + MOM1 * zc0) * inv2;
    float e1  = (MOM * p1 + MOM1 * zc1) * inv2;
    float c0  = init ? e0 : zc0;
    float c1  = init ? e1 : zc1;
    float r0  = present ? c0 : p0;
    float r1  = present ? c1 : p1;
    if (valid) {
      out_protos[(size_t)c * DIMS + k + koff + 0] = r0;
      out_protos[(size_t)c * DIMS + k + koff + 1] = r1;
    }
  }
  if (valid && lane < 16) out_mask[c] = (init || present) ? 1.0f : 0.0f;
}

// ---------------------------------------------------------------------------
extern "C" void kernel_launch(void* const* d_in, const int* in_sizes, int n_in,
                              void* d_out, int out_size, void* d_ws, size_t ws_size,
                              hipStream_t stream) {
  const float*         z     = (const float*)d_in[0];
  const int*           y     = (const int*)d_in[1];          // labels (int32)
  const float*         prot  = (const float*)d_in[2];
  const unsigned char* imask = (const unsigned char*)d_in[3]; // bool (1 byte)

  float* out   = (float*)d_out;                 // [C*D] protos, then [C] mask
  float* seg   = (float*)d_ws;                  // [C, D] segment sums
  float* cnt   = seg + (size_t)C_CLASSES * DIMS; // [C] counts
  const int n  = in_sizes[0] / DIMS;

  (void)n_in; (void)out_size; (void)ws_size;

  hipMemsetAsync(d_ws, 0,
                 (size_t)(C_CLASSES * DIMS + C_CLASSES) * sizeof(float), stream);

  seg_accum_kernel<<<SLICES * PARTS, TPB, 0, stream>>>(z, y, seg, cnt, n);

  finalize_kernel<<<8, 256, 0, stream>>>(seg, cnt, prot, imask,
                                         out, out + (size_t)C_CLASSES * DIMS);
}